// CrossModalCenterLoss_1726576857091
// MI455X (gfx1250) — compile-verified
//
#include <hip/hip_runtime.h>

typedef __attribute__((ext_vector_type(2))) float v2f;
typedef __attribute__((ext_vector_type(8))) float v8f;

#define FEAT   512
#define BATCH  4096
#define NCLS   10000
#define KSPLIT 8                      // K chunks per 16-row tile
#define KCHUNK (FEAT / KSPLIT)        // 64 elements -> 16 WMMA steps per wave

// One wave32 per (16-row tile, 64-element K-chunk). Computes the diagonal of
// the partial product X_tile * C_gathered^T over its K-chunk with
// V_WMMA_F32_16X16X4_F32 (full fp32), partial norms in VALU, and writes 16
// per-row partial distances (clip deferred, it is applied after the K-sum).
__global__ __launch_bounds__(256) void center_dist_kernel(
    const float* __restrict__ x,
    const int*   __restrict__ labels,
    const float* __restrict__ centers,
    float*       __restrict__ partial)   // [BATCH][KSPLIT]
{
    const int lane      = threadIdx.x & 31;
    const int wave      = (blockIdx.x * blockDim.x + threadIdx.x) >> 5;   // 0..2047
    const int tile      = wave >> 3;                                      // 0..255
    const int chunk     = wave & (KSPLIT - 1);                            // 0..7
    const int rowInTile = lane & 15;          // M (for A) == N (for B) for this lane
    const int half      = lane >> 4;          // K sub-pair selector
    const int row       = tile * 16 + rowInTile;
    const int kbase     = chunk * KCHUNK;

    const int lab = labels[row];
    const float* __restrict__ xr = x       + (size_t)row * FEAT;
    const float* __restrict__ cr = centers + (size_t)lab * FEAT;

    v8f acc = {0.f, 0.f, 0.f, 0.f, 0.f, 0.f, 0.f, 0.f};
    float xsq = 0.f, csq = 0.f;

    // A 16x4 layout: lanes 0-15 hold (M=lane, K=k0..k0+1), lanes 16-31 (M=lane-16, K=k0+2..k0+3)
    // B 4x16 layout: lanes 0-15 hold (N=lane, K=k0..k0+1), lanes 16-31 (N=lane-16, K=k0+2..k0+3)
    // -> identical per-lane addressing for both operand streams.
    #pragma unroll 4
    for (int k0 = kbase; k0 < kbase + KCHUNK; k0 += 4) {
        const int k = k0 + 2 * half;
        v2f a = *(const v2f*)(xr + k);
        v2f b = *(const v2f*)(cr + k);
        acc = __builtin_amdgcn_wmma_f32_16x16x4_f32(
                  /*neg_a=*/false, a, /*neg_b=*/false, b,
                  /*c_mod=*/(short)0, acc, /*reuse_a=*/false, /*reuse_b=*/false);
        xsq = fmaf(a.x, a.x, fmaf(a.y, a.y, xsq));
        csq = fmaf(b.x, b.x, fmaf(b.y, b.y, csq));
    }

    // Each lane only summed k%4 in {2h, 2h+1}: fold the two K-halves per row.
    xsq += __shfl_xor(xsq, 16);
    csq += __shfl_xor(csq, 16);

    // Diagonal of the 16x16 f32 D tile:
    //   m <  8 : VGPR m,   lane m        -> lanes 0..7
    //   m >= 8 : VGPR m-8, lane m+16     -> lanes 24..31
    const int  s      = (lane < 8) ? lane : (lane - 24);
    const bool active = (lane < 8) || (lane >= 24);

    float dsel = acc[0];
    if (s == 1) dsel = acc[1];
    if (s == 2) dsel = acc[2];
    if (s == 3) dsel = acc[3];
    if (s == 4) dsel = acc[4];
    if (s == 5) dsel = acc[5];
    if (s == 6) dsel = acc[6];
    if (s == 7) dsel = acc[7];

    // Partial (over this K-chunk) of  ||x||^2 + ||c||^2 - 2<x,c>  -- linear in K.
    const float p = xsq + csq - 2.0f * dsel;

    if (active)
        partial[(size_t)row * KSPLIT + chunk] = p;
}

// Single-block deterministic reduction: per-row K-chunk sum, clip, global sum.
__global__ __launch_bounds__(1024) void center_loss_reduce_kernel(
    const float* __restrict__ partial, float* __restrict__ out)
{
    __shared__ float red[32];
    const int tid  = threadIdx.x;
    const int lane = tid & 31;
    const int wid  = tid >> 5;

    float s = 0.f;
    for (int r = tid; r < BATCH; r += 1024) {
        float d = 0.f;
        #pragma unroll
        for (int j = 0; j < KSPLIT; ++j)
            d += partial[(size_t)r * KSPLIT + j];
        d = fminf(fmaxf(d, 1e-12f), 1e12f);      // clip(dist, 1e-12, 1e12)
        s += d;
    }

    #pragma unroll
    for (int off = 16; off > 0; off >>= 1)
        s += __shfl_xor(s, off);
    if (lane == 0) red[wid] = s;
    __syncthreads();

    if (wid == 0) {
        s = red[lane];
        #pragma unroll
        for (int off = 16; off > 0; off >>= 1)
            s += __shfl_xor(s, off);
        if (lane == 0)
            out[0] = s / (float)BATCH + (float)(NCLS - 1) * 1e-12f;
    }
}

extern "C" void kernel_launch(void* const* d_in, const int* in_sizes, int n_in,
                              void* d_out, int out_size, void* d_ws, size_t ws_size,
                              hipStream_t stream) {
    const float* x       = (const float*)d_in[0];   // [4096, 512] f32
    const int*   labels  = (const int*)  d_in[1];   // [4096] int
    const float* centers = (const float*)d_in[2];   // [10000, 512] f32
    float*       out     = (float*)d_out;           // scalar f32
    float*       partial = (float*)d_ws;            // 4096*8 floats = 128 KB scratch

    const int tiles      = BATCH / 16;              // 256
    const int totalWaves = tiles * KSPLIT;          // 2048 waves
    const int blocks     = totalWaves / 8;          // 256 threads = 8 waves per block

    center_dist_kernel<<<dim3(blocks), dim3(256), 0, stream>>>(x, labels, centers, partial);
    center_loss_reduce_kernel<<<dim3(1), dim3(1024), 0, stream>>>(partial, out);
}